// ErrorAwareEdgeLoss_816043786441
// MI455X (gfx1250) — compile-verified
//
#include <hip/hip_runtime.h>

typedef float v2f __attribute__((ext_vector_type(2)));
typedef float v8f __attribute__((ext_vector_type(8)));

#define BATCH 64
#define NN    256
#define NE    8192

// ---------------------------------------------------------------------------
// Kernel 1: Q[b] = P[b] @ D  using V_WMMA_F32_16X16X4_F32 (fp32 matrix pipe).
// One wave computes a 64x16 strip of Q: 4 accumulator tiles (tile_m) sharing a
// single B fragment per K-step.  K advances 4 per WMMA -> 64 steps, 4 wmma/step.
// Grid: BATCH * 16 tile_n * 4 m-strips = 4096 waves = 512 blocks x 256 threads.
// ---------------------------------------------------------------------------
__global__ __launch_bounds__(256) void qgemm_wmma_f32(
    const float* __restrict__ P, const float* __restrict__ D,
    float* __restrict__ Q) {
  const int wave = (int)((blockIdx.x * blockDim.x + threadIdx.x) >> 5);
  const int lane = (int)(threadIdx.x & 31);

  const int b  = wave >> 6;        // 64 waves per batch
  const int r  = wave & 63;
  const int tn = r >> 2;           // tile_n 0..15
  const int ms = r & 3;            // m-strip 0..3
  const int mbase = ms * 64;
  const int ncol  = tn * 16;

  const float* __restrict__ Pb = P + (size_t)b * NN * NN;
  float* __restrict__ Qb       = Q + (size_t)b * NN * NN;

  // A-fragment addressing (32-bit A 16x4): lane L -> M = L&15;
  // VGPR0 holds K = (L<16 ? 0 : 2), VGPR1 holds K+1.
  const int mloc  = lane & 15;
  const int khalf = (lane >> 4) << 1;     // 0 or 2

  v8f c[4];
  c[0] = (v8f){}; c[1] = (v8f){}; c[2] = (v8f){}; c[3] = (v8f){};

  const float* arow0 = Pb + (size_t)(mbase +  0 + mloc) * NN + khalf;
  const float* arow1 = Pb + (size_t)(mbase + 16 + mloc) * NN + khalf;
  const float* arow2 = Pb + (size_t)(mbase + 32 + mloc) * NN + khalf;
  const float* arow3 = Pb + (size_t)(mbase + 48 + mloc) * NN + khalf;

  for (int k0 = 0; k0 < NN; k0 += 4) {
    v2f a0 = *(const v2f*)(arow0 + k0);
    v2f a1 = *(const v2f*)(arow1 + k0);
    v2f a2 = *(const v2f*)(arow2 + k0);
    v2f a3 = *(const v2f*)(arow3 + k0);
    // B-fragment (4x16, K-major rows striped across lanes):
    // VGPR r holds K = r + (L<16 ? 0 : 2), N = L&15.
    v2f bf;
    bf.x = D[(size_t)(k0 + khalf + 0) * NN + ncol + mloc];
    bf.y = D[(size_t)(k0 + khalf + 1) * NN + ncol + mloc];

    c[0] = __builtin_amdgcn_wmma_f32_16x16x4_f32(false, a0, false, bf,
                                                 (short)0, c[0], false, false);
    c[1] = __builtin_amdgcn_wmma_f32_16x16x4_f32(false, a1, false, bf,
                                                 (short)0, c[1], false, false);
    c[2] = __builtin_amdgcn_wmma_f32_16x16x4_f32(false, a2, false, bf,
                                                 (short)0, c[2], false, false);
    c[3] = __builtin_amdgcn_wmma_f32_16x16x4_f32(false, a3, false, bf,
                                                 (short)0, c[3], false, false);
  }

  // C/D layout: VGPR rr holds M = rr (lanes 0-15) or M = rr+8 (lanes 16-31),
  // N = lane&15.
  const int nloc   = lane & 15;
  const int rowoff = (lane >> 4) * 8;
#pragma unroll
  for (int t = 0; t < 4; ++t) {
#pragma unroll
    for (int rr = 0; rr < 8; ++rr) {
      Qb[(size_t)(mbase + t * 16 + rowoff + rr) * NN + ncol + nloc] = c[t][rr];
    }
  }
}

// ---------------------------------------------------------------------------
// Kernel 2: wave-per-edge gather + dot.  Each block: one batch slice of 256
// edges (8 waves x 32 edges).  Coalesced b128 row loads, wave32 shuffle
// reduction, deterministic per-block partials (no float atomics).
// Grid: BATCH * (NE/256) = 64*32 = 2048 blocks x 256 threads.
// ---------------------------------------------------------------------------
__global__ __launch_bounds__(256) void edge_reduce(
    const float* __restrict__ Q, const float* __restrict__ P,
    const int* __restrict__ ei, const int* __restrict__ ej,
    const float* __restrict__ ew,
    float* __restrict__ loss_part, float* __restrict__ w_part) {
  __shared__ float sl[8];
  __shared__ float sw[8];

  const int b    = (int)(blockIdx.x >> 5);   // 32 blocks per batch
  const int blk  = (int)(blockIdx.x & 31);
  const int wid  = (int)(threadIdx.x >> 5);
  const int lane = (int)(threadIdx.x & 31);
  const int ebase = blk * 256 + wid * 32;

  const float* __restrict__ Pb = P + (size_t)b * NN * NN;
  const float* __restrict__ Qb = Q + (size_t)b * NN * NN;

  float lsum = 0.0f, wsum = 0.0f;

  for (int t = 0; t < 32; ++t) {
    const int e = b * NE + ebase + t;
    const int i = ei[e];
    const int j = ej[e];
    const float w = ew[e];

    const float4* __restrict__ qi = (const float4*)(Qb + (size_t)i * NN);
    const float4* __restrict__ pj = (const float4*)(Pb + (size_t)j * NN);

    float acc = 0.0f;
#pragma unroll
    for (int kk = 0; kk < 2; ++kk) {
      float4 q = qi[kk * 32 + lane];
      float4 p = pj[kk * 32 + lane];
      acc = fmaf(q.x, p.x, fmaf(q.y, p.y, fmaf(q.z, p.z, fmaf(q.w, p.w, acc))));
    }
#pragma unroll
    for (int off = 16; off > 0; off >>= 1) acc += __shfl_xor(acc, off, 32);

    if (lane == 0) {
      lsum = fmaf(w, acc, lsum);
      wsum += w;
    }
  }

  if (lane == 0) { sl[wid] = lsum; sw[wid] = wsum; }
  __syncthreads();
  if (threadIdx.x == 0) {
    float L = 0.0f, W = 0.0f;
#pragma unroll
    for (int k = 0; k < 8; ++k) { L += sl[k]; W += sw[k]; }
    loss_part[blockIdx.x] = L;
    w_part[blockIdx.x]    = W;
  }
}

// ---------------------------------------------------------------------------
// Kernel 3: per-batch loss/max(w,1e-8), then mean over batch.  1 block x 64.
// ---------------------------------------------------------------------------
__global__ __launch_bounds__(64) void finalize(
    const float* __restrict__ loss_part, const float* __restrict__ w_part,
    float* __restrict__ out) {
  __shared__ float s[64];
  const int b = (int)threadIdx.x;
  float L = 0.0f, W = 0.0f;
  for (int k = 0; k < 32; ++k) {
    L += loss_part[b * 32 + k];
    W += w_part[b * 32 + k];
  }
  s[b] = L / fmaxf(W, 1e-8f);
  __syncthreads();
  for (int off = 32; off > 0; off >>= 1) {
    if (b < off) s[b] += s[b + off];
    __syncthreads();
  }
  if (b == 0) out[0] = s[0] / (float)BATCH;
}

extern "C" void kernel_launch(void* const* d_in, const int* in_sizes, int n_in,
                              void* d_out, int out_size, void* d_ws, size_t ws_size,
                              hipStream_t stream) {
  const float* P  = (const float*)d_in[0];   // [B, N, N]
  const float* D  = (const float*)d_in[1];   // [N, N]
  const int* ei   = (const int*)d_in[2];     // [B, E]
  const int* ej   = (const int*)d_in[3];     // [B, E]
  const float* ew = (const float*)d_in[4];   // [B, E]
  float* out = (float*)d_out;

  float* Q         = (float*)d_ws;                                   // B*N*N
  float* loss_part = Q + (size_t)BATCH * NN * NN;                    // 2048
  float* w_part    = loss_part + (size_t)BATCH * (NE / 256);         // 2048

  // Kernel 1: 4096 waves -> 512 blocks of 256 threads
  qgemm_wmma_f32<<<512, 256, 0, stream>>>(P, D, Q);
  // Kernel 2: 2048 blocks of 256 threads
  edge_reduce<<<BATCH * (NE / 256), 256, 0, stream>>>(Q, P, ei, ej, ew,
                                                      loss_part, w_part);
  // Kernel 3
  finalize<<<1, 64, 0, stream>>>(loss_part, w_part, out);
}